// Link_83047487635827
// MI455X (gfx1250) — compile-verified
//
#include <hip/hip_runtime.h>

typedef float v2f __attribute__((ext_vector_type(2)));
typedef float v8f __attribute__((ext_vector_type(8)));

static constexpr int TT = 128;     // tags (GEMM M)
static constexpr int NT = 32768;   // tokens (GEMM K)
static constexpr int DD = 1024;    // feature dim (GEMM N)
static constexpr int KCHUNK = 2048;
static constexpr int KSPLIT = NT / KCHUNK;   // 16
static constexpr int DSLICE = 128;

// workspace layout (bytes)
static constexpr size_t ACC_OFF = 0;                                // TT*DD f32 = 512 KB
static constexpr size_t CNT_OFF = (size_t)TT * DD * sizeof(float);  // TT i32
static constexpr size_t OWN_OFF = CNT_OFF + 4096;                   // NT i32 = 128 KB

// ---------------- kernel 0: zero the accumulator + counts ----------------
__global__ void zero_kernel(float* __restrict__ acc, int* __restrict__ cnt) {
    int i = blockIdx.x * blockDim.x + threadIdx.x;
    if (i < TT * DD) acc[i] = 0.0f;
    if (i < TT)      cnt[i] = 0;
}

// ---------------- kernel 1: per-token owner tag + row counts ----------------
// temp[i][n] is one-hot at the max i with tag_to_token[i][n] > 0 (math identity
// of the clamp(binary - suffix_sum) expression), so t2t reduces to own[] + 1/count.
__global__ void owner_kernel(const float* __restrict__ t2t,
                             int* __restrict__ own, int* __restrict__ cnt) {
    int n = blockIdx.x * blockDim.x + threadIdx.x;
    if (n >= NT) return;
    int o = -1;
    for (int i = 0; i < TT; ++i) {
        float v = t2t[(size_t)i * NT + n];     // coalesced across n each step
        if (v > 0.0f) o = i;
    }
    own[n] = o;
    if (o >= 0) atomicAdd(&cnt[o], 1);
}

// ---------------- kernel 2: WMMA f32 16x16x4 GEMM  acc += t2t @ inputs ----------------
// One wave per workgroup. Grid: (DD/16 d-tiles) x (KSPLIT k-chunks).
// A (t2t) synthesized on the fly from own[]/inv[]. own[k..k+3] are loaded at uniform
// addresses -> SGPRs; a uniform active-subtile mask skips the (mostly all-zero-A) WMMAs
// via scalar branches, which leave EXEC all-ones as WMMA requires.
__global__ __launch_bounds__(32)
void gemm_wmma_kernel(const float* __restrict__ inp, const int* __restrict__ own,
                      const int* __restrict__ cnt, float* __restrict__ acc) {
    const int lane  = threadIdx.x;        // 0..31
    const int laneN = lane & 15;
    const int hi    = lane >> 4;          // 0: A/B K-pair 0/1, 1: K-pair 2/3
    const int dbase = blockIdx.x * 16;
    const int kbeg  = blockIdx.y * KCHUNK;

    // per-lane row scale: inv[mb] = 1/count for GEMM row (mb*16 + laneN)
    float inv[8];
#pragma unroll
    for (int mb = 0; mb < 8; ++mb) {
        int c = cnt[mb * 16 + laneN];
        inv[mb] = (c > 0) ? (1.0f / (float)c) : 0.0f;
    }

    v8f accv[8];
#pragma unroll
    for (int mb = 0; mb < 8; ++mb)
#pragma unroll
        for (int v = 0; v < 8; ++v) accv[mb][v] = 0.0f;

    unsigned dirty = 0;                   // uniform: which M-subtiles were touched

    for (int k = kbeg; k < kbeg + KCHUNK; k += 4) {
        // prefetch ahead (speculative; safely dropped past buffer end per ISA)
        int pr = k + 64; if (pr >= NT) pr = NT - 4;
        __builtin_prefetch(&inp[(size_t)pr * DD + dbase + laneN], 0, 1);

        // owners for K columns k..k+3: uniform addresses -> scalar loads
        const int o0 = own[k + 0];
        const int o1 = own[k + 1];
        const int o2 = own[k + 2];
        const int o3 = own[k + 3];
        unsigned mask = 0;
        if (o0 >= 0) mask |= 1u << (o0 >> 4);
        if (o1 >= 0) mask |= 1u << (o1 >> 4);
        if (o2 >= 0) mask |= 1u << (o2 >> 4);
        if (o3 >= 0) mask |= 1u << (o3 >> 4);
        if (mask == 0) continue;          // whole 4-column slab contributes nothing
        dirty |= mask;

        // B fragment: 4x16 tile of inputs rows k..k+3, cols dbase..+15
        // layout: VGPR0 = row k+2*hi, VGPR1 = row k+2*hi+1, N = laneN
        const int r0 = k + hi * 2;
        v2f b;
        b.x = inp[(size_t)r0 * DD + dbase + laneN];
        b.y = inp[(size_t)(r0 + 1) * DD + dbase + laneN];

        // per-lane owner values matching the A K-slots this lane holds
        const int oa = hi ? o2 : o0;
        const int ob = hi ? o3 : o1;

#pragma unroll
        for (int mb = 0; mb < 8; ++mb) {
            if (mask & (1u << mb)) {      // uniform branch (EXEC untouched)
                const int m = mb * 16 + laneN;       // A-row this lane holds
                v2f a;
                a.x = (oa == m) ? inv[mb] : 0.0f;    // t2t[m][k + 2*hi]
                a.y = (ob == m) ? inv[mb] : 0.0f;    // t2t[m][k + 2*hi + 1]
                accv[mb] = __builtin_amdgcn_wmma_f32_16x16x4_f32(
                    false, a, false, b, (short)0, accv[mb], false, false);
            }
        }
    }

    // flush only touched subtiles: C/D element (mb*16 + v + hi*8, dbase + laneN)
#pragma unroll
    for (int mb = 0; mb < 8; ++mb) {
        if (dirty & (1u << mb)) {
#pragma unroll
            for (int v = 0; v < 8; ++v) {
                int m = mb * 16 + v + hi * 8;
                __hip_atomic_fetch_add(&acc[(size_t)m * DD + dbase + laneN], accv[mb][v],
                                       __ATOMIC_RELAXED, __HIP_MEMORY_SCOPE_AGENT);
            }
        }
    }
}

// ---------------- kernel 3: triangular recurrence ----------------
// child == binarize(gat_mask): in the reference, rows j>i are still zero when the
// inner update reads them (rows are assigned at outer iteration == their index,
// ascending), so the subtraction is always a no-op. D columns are independent:
// one thread per column, column slab in LDS, i descending; child[i][i]=1 reads the
// pre-update self row exactly as the reference einsum does.
__global__ __launch_bounds__(DSLICE)
void rec_kernel(const int* __restrict__ gm, const float* __restrict__ acc,
                float* __restrict__ out) {
    extern __shared__ float slab[];              // TT x DSLICE
    const int d = threadIdx.x;                   // 0..DSLICE-1
    const int dbase = blockIdx.x * DSLICE;

    for (int r = 0; r < TT; ++r)
        slab[r * DSLICE + d] = acc[(size_t)r * DD + dbase + d];

    for (int i = TT - 1; i >= 0; --i) {
        float s = 0.0f;
        for (int t = 0; t < TT; ++t) {
            // gm[i*TT+t] is wave-uniform -> scalar loads
            float c = (gm[i * TT + t] != 0) ? 1.0f : 0.0f;
            s += c * slab[t * DSLICE + d];
        }
        slab[i * DSLICE + d] = s;                // column-private, no barrier needed
    }

    for (int r = 0; r < TT; ++r)
        out[(size_t)r * DD + dbase + d] = slab[r * DSLICE + d];
}

// ---------------- launch ----------------
extern "C" void kernel_launch(void* const* d_in, const int* in_sizes, int n_in,
                              void* d_out, int out_size, void* d_ws, size_t ws_size,
                              hipStream_t stream) {
    const float* inputs = (const float*)d_in[0];   // (1, N, D) f32
    const float* t2t    = (const float*)d_in[1];   // (1, T, N) f32
    const int*   gm     = (const int*)d_in[2];     // (1, T, T) i32
    float* out = (float*)d_out;                    // (1, T, D) f32

    char* ws  = (char*)d_ws;
    float* acc = (float*)(ws + ACC_OFF);
    int*   cnt = (int*)(ws + CNT_OFF);
    int*   own = (int*)(ws + OWN_OFF);

    zero_kernel<<<(TT * DD + 255) / 256, 256, 0, stream>>>(acc, cnt);
    owner_kernel<<<(NT + 255) / 256, 256, 0, stream>>>(t2t, own, cnt);
    dim3 grid(DD / 16, KSPLIT);
    gemm_wmma_kernel<<<grid, 32, 0, stream>>>(inputs, own, cnt, acc);
    rec_kernel<<<DD / DSLICE, DSLICE, TT * DSLICE * sizeof(float), stream>>>(gm, acc, out);
}